// MultiBoxLoss_4672924418145
// MI455X (gfx1250) — compile-verified
//
#include <hip/hip_runtime.h>
#include <hip/hip_bf16.h>
#include <math.h>

// ---------------- problem constants (match reference) ----------------
#define Bn 64
#define Pn 8732
#define Cn 81
#define On 81
#define THRESH 0.5f
#define NEGPOS 3

#define NB1 35              // ceil(Pn/256)
#define TILE5 64            // priors per CE tile
#define NB5 137             // ceil(Pn/TILE5)

// ---------------- vector types ----------------
typedef float v2f  __attribute__((ext_vector_type(2)));
typedef float v8f  __attribute__((ext_vector_type(8)));
typedef unsigned int u32x4 __attribute__((ext_vector_type(4)));
typedef int   i32x4 __attribute__((ext_vector_type(4)));
typedef int   i32x8 __attribute__((ext_vector_type(8)));

// ---------------- TDM 1-D global->LDS copy (CDNA5 Tensor Data Mover) ----
// Builds a D# per CDNA5 ISA §8: group0 = {count/lds_addr/global_addr/type},
// group1 = {data_size, tensor_dim0 (valid elems; OOB reads return 0),
// tile_dim0, strides}. 4-byte elements. Issued once per wave (EXEC-agnostic).
__device__ __forceinline__ void tdm_load_1d(const void* gsrc, unsigned lds_off,
                                            unsigned valid_elems, unsigned tile_elems) {
  unsigned long long ga = (unsigned long long)(size_t)gsrc;
  u32x4 g0;
  g0[0] = 1u;                                              // count=1, user desc
  g0[1] = lds_off;                                         // lds_addr
  g0[2] = (unsigned)(ga & 0xffffffffu);                    // global_addr[31:0]
  g0[3] = (unsigned)((ga >> 32) & 0x01ffffffu) | (2u << 30); // addr[56:32] | type=2
  unsigned td0 = valid_elems;                              // tensor_dim0
  unsigned td1 = 1u;                                       // tensor_dim1
  unsigned long long t0s = tile_elems;                     // tensor_dim0_stride
  unsigned long long t1s = 0ull;
  i32x8 g1;
  g1[0] = (int)(2u << 16);                                 // data_size=2 (4 bytes)
  g1[1] = (int)((td0 & 0xffffu) << 16);                    // tensor_dim0[15:0]
  g1[2] = (int)(((td0 >> 16) & 0xffffu) | ((td1 & 0xffffu) << 16));
  g1[3] = (int)(((td1 >> 16) & 0xffffu) | ((tile_elems & 0xffffu) << 16)); // tile_dim0
  g1[4] = 0;                                               // tile_dim1=0, tile_dim2=0
  g1[5] = (int)(t0s & 0xffffffffu);
  g1[6] = (int)(((t0s >> 32) & 0xffffu) | ((t1s & 0xffffu) << 16));
  g1[7] = (int)((t1s >> 16) & 0xffffffffu);
  i32x4 gz = {0, 0, 0, 0};
#if defined(__clang_major__) && (__clang_major__ >= 23)
  i32x8 gz8 = {0, 0, 0, 0, 0, 0, 0, 0};
  __builtin_amdgcn_tensor_load_to_lds(g0, g1, gz, gz, gz8, 0);
#else
  __builtin_amdgcn_tensor_load_to_lds(g0, g1, gz, gz, 0);
#endif
}

// ---------------- K0: priors cxcywh -> xyxy, zero n_pos -----------------
__global__ void k_priors(const float* __restrict__ pr, float* __restrict__ pxy,
                         int* __restrict__ npos) {
  int t = blockIdx.x * blockDim.x + threadIdx.x;
  if (t < Bn) npos[t] = 0;
  if (t < Pn) {
    float cx = pr[t * 4 + 0], cy = pr[t * 4 + 1];
    float w = pr[t * 4 + 2], h = pr[t * 4 + 3];
    pxy[t * 4 + 0] = cx - w * 0.5f;
    pxy[t * 4 + 1] = cy - h * 0.5f;
    pxy[t * 4 + 2] = cx + w * 0.5f;
    pxy[t * 4 + 3] = cy + h * 0.5f;
  }
}

// ---------------- K1: per-prior argmax over objects (first-max ties) ----
__global__ void k_match_prior(const float* __restrict__ boxes, const float* __restrict__ pxy,
                              float* __restrict__ ov, int* __restrict__ obj) {
  int b = blockIdx.y;
  __shared__ float bx[On * 4];
  __shared__ float ar[On];
  for (int i = threadIdx.x; i < On * 4; i += blockDim.x) bx[i] = boxes[b * On * 4 + i];
  __syncthreads();
  for (int i = threadIdx.x; i < On; i += blockDim.x)
    ar[i] = (bx[i * 4 + 2] - bx[i * 4 + 0]) * (bx[i * 4 + 3] - bx[i * 4 + 1]);
  __syncthreads();
  int p = blockIdx.x * blockDim.x + threadIdx.x;
  if (p >= Pn) return;
  float px1 = pxy[p * 4 + 0], py1 = pxy[p * 4 + 1];
  float px2 = pxy[p * 4 + 2], py2 = pxy[p * 4 + 3];
  float pa = (px2 - px1) * (py2 - py1);
  float best = -1.0f; int bi = 0;
  for (int o = 0; o < On; ++o) {
    float lx = fmaxf(px1, bx[o * 4 + 0]), ly = fmaxf(py1, bx[o * 4 + 1]);
    float ux = fminf(px2, bx[o * 4 + 2]), uy = fminf(py2, bx[o * 4 + 3]);
    float w = fmaxf(ux - lx, 0.0f), h = fmaxf(uy - ly, 0.0f);
    float inter = w * h;
    float iou = inter / (ar[o] + pa - inter);
    if (iou > best) { best = iou; bi = o; }
  }
  ov[(size_t)b * Pn + p] = best;
  obj[(size_t)b * Pn + p] = bi;
}

// ---------------- K2: per-object argmax over priors (smallest-index ties)
__global__ void k_match_obj(const float* __restrict__ boxes, const float* __restrict__ pxy,
                            int* __restrict__ pfo) {
  int b = blockIdx.y, o = blockIdx.x;
  const float* bo = boxes + ((size_t)b * On + o) * 4;
  float bx1 = bo[0], by1 = bo[1], bx2 = bo[2], by2 = bo[3];
  float ba = (bx2 - bx1) * (by2 - by1);
  float best = -1.0f; int bi = 0;
  for (int p = threadIdx.x; p < Pn; p += blockDim.x) {
    float lx = fmaxf(bx1, pxy[p * 4 + 0]), ly = fmaxf(by1, pxy[p * 4 + 1]);
    float ux = fminf(bx2, pxy[p * 4 + 2]), uy = fminf(by2, pxy[p * 4 + 3]);
    float w = fmaxf(ux - lx, 0.0f), h = fmaxf(uy - ly, 0.0f);
    float inter = w * h;
    float pa = (pxy[p * 4 + 2] - pxy[p * 4 + 0]) * (pxy[p * 4 + 3] - pxy[p * 4 + 1]);
    float iou = inter / (ba + pa - inter);
    if (iou > best) { best = iou; bi = p; }
  }
  __shared__ float sv[256];
  __shared__ int si[256];
  sv[threadIdx.x] = best; si[threadIdx.x] = bi;
  __syncthreads();
  for (int s = 128; s > 0; s >>= 1) {
    if ((int)threadIdx.x < s) {
      float v2 = sv[threadIdx.x + s]; int i2 = si[threadIdx.x + s];
      if (v2 > sv[threadIdx.x] || (v2 == sv[threadIdx.x] && i2 < si[threadIdx.x])) {
        sv[threadIdx.x] = v2; si[threadIdx.x] = i2;
      }
    }
    __syncthreads();
  }
  if (threadIdx.x == 0) pfo[b * On + o] = si[0];
}

// ---------------- K3: override matched priors (sequential, last wins) ---
__global__ void k_override(const int* __restrict__ pfo, float* __restrict__ ov,
                           int* __restrict__ obj) {
  if (threadIdx.x != 0) return;
  int b = blockIdx.x;
  for (int o = 0; o < On; ++o) {
    int p = pfo[b * On + o];
    obj[(size_t)b * Pn + p] = o;
    ov[(size_t)b * Pn + p] = 1.0f;
  }
}

// ---------------- K4: labels, gcxgcy encode, L1 partials, n_pos ---------
__global__ void k_target(const float* __restrict__ boxes, const int* __restrict__ labels,
                         const float* __restrict__ priors, const float* __restrict__ ploc,
                         const float* __restrict__ ov, const int* __restrict__ obj,
                         int* __restrict__ labout, int* __restrict__ npos,
                         float* __restrict__ l1p) {
  int b = blockIdx.y;
  __shared__ float bx[On * 4];
  __shared__ int lb[On];
  __shared__ float sred[256];
  __shared__ int sc[256];
  for (int i = threadIdx.x; i < On * 4; i += blockDim.x) bx[i] = boxes[b * On * 4 + i];
  for (int i = threadIdx.x; i < On; i += blockDim.x) lb[i] = labels[b * On + i];
  __syncthreads();
  int p = blockIdx.x * blockDim.x + threadIdx.x;
  float l1 = 0.0f; int cnt = 0;
  if (p < Pn) {
    int o = obj[(size_t)b * Pn + p];
    float v = ov[(size_t)b * Pn + p];
    int lab = (v < THRESH) ? 0 : lb[o];
    labout[(size_t)b * Pn + p] = lab;
    if (lab != 0) {
      cnt = 1;
      const float* bo = &bx[o * 4];
      float cx = (bo[0] + bo[2]) * 0.5f, cy = (bo[1] + bo[3]) * 0.5f;
      float w = bo[2] - bo[0], h = bo[3] - bo[1];
      float pcx = priors[p * 4 + 0], pcy = priors[p * 4 + 1];
      float pw = priors[p * 4 + 2], ph = priors[p * 4 + 3];
      float g0 = (cx - pcx) / pw * 10.0f;
      float g1 = (cy - pcy) / ph * 10.0f;
      float g2 = logf(w / pw) * 5.0f;
      float g3 = logf(h / ph) * 5.0f;
      const float* pl = &ploc[((size_t)b * Pn + p) * 4];
      l1 = fabsf(pl[0] - g0) + fabsf(pl[1] - g1) + fabsf(pl[2] - g2) + fabsf(pl[3] - g3);
    }
  }
  sred[threadIdx.x] = l1; sc[threadIdx.x] = cnt;
  __syncthreads();
  for (int s = 128; s > 0; s >>= 1) {
    if ((int)threadIdx.x < s) {
      sred[threadIdx.x] += sred[threadIdx.x + s];
      sc[threadIdx.x] += sc[threadIdx.x + s];
    }
    __syncthreads();
  }
  if (threadIdx.x == 0) {
    l1p[b * NB1 + blockIdx.x] = sred[0];
    atomicAdd(&npos[b], sc[0]);            // integer atomic: deterministic
  }
}

// ---------------- K5: CE via TDM-staged tiles (bandwidth-dominant) ------
// Block = 8 waves, TILE5=64 priors. One TDM DMA of 64x81 f32 into LDS,
// s_wait_tensorcnt, then one wave per prior does a 32-lane logsumexp.
__global__ void k_conf(const float* __restrict__ scores, const int* __restrict__ lab,
                       float* __restrict__ neg, float* __restrict__ cep) {
  int b = blockIdx.y;
  int p0 = blockIdx.x * TILE5;
  __shared__ float tile[TILE5 * Cn];   // first LDS object -> offset 0
  __shared__ float posc[TILE5];
  int nvalid = Pn - p0; if (nvalid > TILE5) nvalid = TILE5;
  if (threadIdx.x < 32) {              // wave 0 issues the DMA + waits
    const float* src = scores + ((size_t)b * Pn + p0) * Cn;
    tdm_load_1d(src, 0u, (unsigned)(nvalid * Cn), (unsigned)(TILE5 * Cn));
    __builtin_amdgcn_s_wait_tensorcnt(0);
  }
  __syncthreads();
  int wave = threadIdx.x >> 5, lane = threadIdx.x & 31;
  for (int i = 0; i < TILE5 / 8; ++i) {
    int l = wave * 8 + i;
    int p = p0 + l;
    if (p < Pn) {
      const float* row = &tile[l * Cn];
      float x0 = row[lane];
      float x1 = row[lane + 32];
      float x2 = (lane < Cn - 64) ? row[lane + 64] : -INFINITY;
      float m = fmaxf(fmaxf(x0, x1), x2);
      for (int off = 16; off > 0; off >>= 1) m = fmaxf(m, __shfl_xor(m, off, 32));
      float s = expf(x0 - m) + expf(x1 - m) + ((lane < Cn - 64) ? expf(x2 - m) : 0.0f);
      for (int off = 16; off > 0; off >>= 1) s += __shfl_xor(s, off, 32);
      float lse = m + logf(s);
      int L = lab[(size_t)b * Pn + p];
      float conf = lse - row[L];
      if (lane == 0) {
        if (L != 0) { posc[l] = conf; neg[(size_t)b * Pn + p] = 0.0f; }
        else        { posc[l] = 0.0f; neg[(size_t)b * Pn + p] = fmaxf(conf, 0.0f); }
      }
    } else if (lane == 0) {
      posc[l] = 0.0f;
    }
  }
  __syncthreads();
  if (threadIdx.x == 0) {              // fixed-order block partial (deterministic)
    float s = 0.0f;
    for (int i = 0; i < TILE5; ++i) s += posc[i];
    cep[b * NB5 + blockIdx.x] = s;
  }
}

// ---------------- K6: exact top-K sum via radix select -----------------
// All conf_neg >= 0, so float bits are order-monotonic as unsigned.
__global__ void k_hardneg(const float* __restrict__ neg, const int* __restrict__ npos,
                          float* __restrict__ hs) {
  int b = blockIdx.x;
  const float* v = neg + (size_t)b * Pn;
  int K = NEGPOS * npos[b];
  if (K > Pn) K = Pn;
  __shared__ int scnt[256];
  __shared__ float ssum[256];
  __shared__ unsigned s_prefix;
  __shared__ int s_k;
  if (threadIdx.x == 0) { s_prefix = 0u; s_k = K; }
  __syncthreads();
  if (K > 0) {
    for (int bit = 31; bit >= 0; --bit) {
      unsigned prefix = s_prefix;
      int c = 0;
      for (int i = threadIdx.x; i < Pn; i += 256) {
        unsigned u = __float_as_uint(v[i]);
        unsigned hi = (bit < 31) ? (u >> (bit + 1)) : 0u;
        unsigned ph = (bit < 31) ? (prefix >> (bit + 1)) : 0u;
        c += (hi == ph && ((u >> bit) & 1u)) ? 1 : 0;
      }
      scnt[threadIdx.x] = c;
      __syncthreads();
      for (int s = 128; s > 0; s >>= 1) {
        if ((int)threadIdx.x < s) scnt[threadIdx.x] += scnt[threadIdx.x + s];
        __syncthreads();
      }
      if (threadIdx.x == 0) {
        if (scnt[0] >= s_k) s_prefix = prefix | (1u << bit);
        else s_k -= scnt[0];
      }
      __syncthreads();
    }
    unsigned tb = s_prefix;
    float tval = __uint_as_float(tb);
    float sgt = 0.0f; int cgt = 0;
    for (int i = threadIdx.x; i < Pn; i += 256) {
      unsigned u = __float_as_uint(v[i]);
      if (u > tb) { sgt += v[i]; cgt += 1; }
    }
    ssum[threadIdx.x] = sgt; scnt[threadIdx.x] = cgt;
    __syncthreads();
    for (int s = 128; s > 0; s >>= 1) {
      if ((int)threadIdx.x < s) {
        ssum[threadIdx.x] += ssum[threadIdx.x + s];
        scnt[threadIdx.x] += scnt[threadIdx.x + s];
      }
      __syncthreads();
    }
    if (threadIdx.x == 0) hs[b] = ssum[0] + (float)(K - scnt[0]) * tval;
  } else if (threadIdx.x == 0) {
    hs[b] = 0.0f;
  }
}

// ---------------- K7: final scalar via WMMA reductions ------------------
// ones(16x4) x B(4x16): every row of D is the per-column sum of B, so each
// WMMA folds 64 f32 into the accumulator. n MUST be a multiple of 64 (all
// partial arrays are sized that way), so loads are unconditional coalesced
// float2 (b64). Two accumulator chains avoid WMMA->WMMA RAW stalls; merged
// with exact fp32 adds at the end. Deterministic fixed order.
__device__ __forceinline__ float wave_sum_wmma(const float* a, int n) {
  int lane = threadIdx.x & 31;
  v8f acc0 = {};
  v8f acc1 = {};
  v2f ones; ones[0] = 1.0f; ones[1] = 1.0f;
  const v2f* a2 = (const v2f*)a;              // lane holds 2 consecutive floats
  int nchunk = n >> 6;                        // 64 floats per WMMA
  int i = 0;
  for (; i + 1 < nchunk; i += 2) {
    v2f b0 = a2[(i << 5) + lane];
    v2f b1 = a2[((i + 1) << 5) + lane];
    acc0 = __builtin_amdgcn_wmma_f32_16x16x4_f32(false, ones, false, b0,
                                                 (short)0, acc0, false, false);
    acc1 = __builtin_amdgcn_wmma_f32_16x16x4_f32(false, ones, false, b1,
                                                 (short)0, acc1, false, false);
  }
  if (i < nchunk) {
    v2f b0 = a2[(i << 5) + lane];
    acc0 = __builtin_amdgcn_wmma_f32_16x16x4_f32(false, ones, false, b0,
                                                 (short)0, acc0, false, false);
  }
  float x = (lane < 16) ? (acc0[0] + acc1[0]) : 0.0f;   // D row 0, cols 0..15
  for (int off = 16; off > 0; off >>= 1) x += __shfl_xor(x, off, 32);
  return x;
}

__global__ void k_final(const float* __restrict__ l1p, int n1,
                        const float* __restrict__ cep, int n2,
                        const float* __restrict__ hs, const int* __restrict__ npos,
                        float* __restrict__ out) {
  float l1t = wave_sum_wmma(l1p, n1);
  float cpt = wave_sum_wmma(cep, n2);
  float hnt = wave_sum_wmma(hs, Bn);
  if (threadIdx.x == 0) {
    int npt = 0;
    for (int i = 0; i < Bn; ++i) npt += npos[i];
    float nptf = (float)npt;
    float loc_loss = l1t / (nptf * 4.0f);
    float conf_loss = (hnt + cpt) / nptf;
    out[0] = conf_loss + loc_loss;
  }
}

// ---------------- launch ----------------
extern "C" void kernel_launch(void* const* d_in, const int* in_sizes, int n_in,
                              void* d_out, int out_size, void* d_ws, size_t ws_size,
                              hipStream_t stream) {
  const float* ploc   = (const float*)d_in[0];  // [B,P,4]
  const float* scores = (const float*)d_in[1];  // [B,P,C]
  const float* boxes  = (const float*)d_in[2];  // [B,O,4]
  const int*   labels = (const int*)d_in[3];    // [B,O]
  const float* priors = (const float*)d_in[4];  // [P,4]

  // workspace layout (element offsets)
  float* ws = (float*)d_ws;
  size_t o_pxy = 0;                          // P*4 f
  size_t o_ov  = o_pxy + (size_t)Pn * 4;     // B*P f
  size_t o_obj = o_ov + (size_t)Bn * Pn;     // B*P i
  size_t o_pfo = o_obj + (size_t)Bn * Pn;    // B*O i
  size_t o_lab = o_pfo + (size_t)Bn * On;    // B*P i
  size_t o_neg = o_lab + (size_t)Bn * Pn;    // B*P f
  size_t o_np  = o_neg + (size_t)Bn * Pn;    // B   i
  size_t o_l1p = o_np + Bn;                  // B*NB1 f (multiple of 64)
  size_t o_cep = o_l1p + (size_t)Bn * NB1;   // B*NB5 f (multiple of 64)
  size_t o_hs  = o_cep + (size_t)Bn * NB5;   // B   f  (multiple of 64)

  float* pxy = ws + o_pxy;
  float* ov  = ws + o_ov;
  int*   obj = (int*)(ws + o_obj);
  int*   pfo = (int*)(ws + o_pfo);
  int*   lab = (int*)(ws + o_lab);
  float* neg = ws + o_neg;
  int*   np  = (int*)(ws + o_np);
  float* l1p = ws + o_l1p;
  float* cep = ws + o_cep;
  float* hs  = ws + o_hs;

  dim3 b256(256);
  k_priors<<<dim3(NB1), b256, 0, stream>>>(priors, pxy, np);
  k_match_prior<<<dim3(NB1, Bn), b256, 0, stream>>>(boxes, pxy, ov, obj);
  k_match_obj<<<dim3(On, Bn), b256, 0, stream>>>(boxes, pxy, pfo);
  k_override<<<dim3(Bn), dim3(32), 0, stream>>>(pfo, ov, obj);
  k_target<<<dim3(NB1, Bn), b256, 0, stream>>>(boxes, labels, priors, ploc, ov, obj,
                                               lab, np, l1p);
  k_conf<<<dim3(NB5, Bn), b256, 0, stream>>>(scores, lab, neg, cep);
  k_hardneg<<<dim3(Bn), b256, 0, stream>>>(neg, np, hs);
  k_final<<<dim3(1), dim3(32), 0, stream>>>(l1p, Bn * NB1, cep, Bn * NB5, hs, np,
                                            (float*)d_out);
}